// Upsample_17961553232405
// MI455X (gfx1250) — compile-verified
//
#include <hip/hip_runtime.h>

// k-NN (k=4) upsampling for MI455X / gfx1250, wave32.
//
//   d2[m][n] = |c_m|^2 + |q_n|^2 - 2 c_m.q_n  as a K=4 dot product:
//     A row (candidate m) = (cx, cy, |c|^2, 1)
//     B col (query n)     = (-2qx, -2qy, 1, |q|^2)
//   -> one V_WMMA_F32_16X16X4_F32 per 16(cand) x 16(query) distance tile.
//   Top-4 kept as u32 keys: [f32 distance bits & ~0x1FFF] | candidate_idx.
//   Clamp-below-0 via single v_med3_f32; A operand software-pipelined from LDS.

typedef float v2f __attribute__((ext_vector_type(2)));
typedef float v8f __attribute__((ext_vector_type(8)));

#define N_IN   8192
#define N_NEW  (N_IN * 3)
#define N_TOT  (N_IN + N_NEW)      // 32768
#define C_CH   64
#define NTILES (N_IN / 16)         // 512 candidate tiles
#define QPB    128                 // queries per block (8 waves x 16)
#define THREADS 256
#define FLT_BIG 3.0e38f

// Dynamic LDS layout:
//   [0, NTILES*32*8)         : A-matrix staging, v2f per (tile, lane-slot)
//   [NTILES*32*8, +QPB*4*4)  : top-4 indices per block-local query
#define LDS_A_BYTES   (NTILES * 32 * 8)          // 131072
#define LDS_IDX_BYTES (QPB * 4 * 4)              // 2048
#define LDS_BYTES     (LDS_A_BYTES + LDS_IDX_BYTES)

__device__ __forceinline__ unsigned umin32(unsigned a, unsigned b) { return a < b ? a : b; }
__device__ __forceinline__ unsigned umax32(unsigned a, unsigned b) { return a > b ? a : b; }

// insert key x into ascending sorted (s0<=s1<=s2<=s3), keep 4 smallest
#define TOP4_INSERT(x_)                                              \
  do {                                                               \
    unsigned x__ = (x_), t__;                                        \
    t__ = umin32(s0, x__); x__ = umax32(s0, x__); s0 = t__;          \
    t__ = umin32(s1, x__); x__ = umax32(s1, x__); s1 = t__;          \
    t__ = umin32(s2, x__); x__ = umax32(s2, x__); s2 = t__;          \
    s3 = umin32(s3, x__);                                            \
  } while (0)

extern "C" __global__ __launch_bounds__(THREADS)
void knn_upsample_wmma(const float* __restrict__ values,      // [64, 8192]
                       const float* __restrict__ coords,      // [8192, 2]
                       const float* __restrict__ new_coords,  // [24576, 2]
                       const float* __restrict__ shift,       // [2]
                       float* __restrict__ out)               // [64, 32768]
{
    extern __shared__ char smem_raw[];
    v2f* aStage = (v2f*)smem_raw;                      // NTILES*32 entries
    int* idxBuf = (int*)(smem_raw + LDS_A_BYTES);      // QPB*4 ints

    const int tid  = threadIdx.x;
    const int lane = tid & 31;
    const int wave = tid >> 5;
    const float sx = shift[0];
    const float sy = shift[1];

    // ---------------- Phase 1: stage candidate operands in WMMA A-layout ----
    // slot < 16 : (cx, cy) of m = tile*16 + slot           (A VGPR0/1, K=0,1)
    // slot >= 16: (|c|^2, 1.0) of m = tile*16 + slot - 16  (A VGPR0/1, K=2,3)
    for (int e = tid; e < NTILES * 32; e += THREADS) {
        const int slot = e & 31;
        const int m    = (e >> 5) * 16 + (slot & 15);
        const float cx = coords[2 * m + 0];
        const float cy = coords[2 * m + 1];
        v2f v;
        if (slot < 16) { v.x = cx;                v.y = cy;   }
        else           { v.x = cx * cx + cy * cy; v.y = 1.0f; }
        aStage[e] = v;
    }
    __syncthreads();

    // ---------------- Phase 2: WMMA distance tiles + packed-key top-4 -------
    const int qbase = blockIdx.x * QPB;
    const int n = qbase + wave * 16 + (lane & 15);     // this lane's query column
    float qx, qy;
    if (n < N_IN) { qx = coords[2 * n + 0];              qy = coords[2 * n + 1]; }
    else          { qx = new_coords[2 * (n - N_IN) + 0]; qy = new_coords[2 * (n - N_IN) + 1]; }
    qx -= sx;  qy -= sy;

    v2f b;  // B operand: lanes 0-15 hold K=0,1; lanes 16-31 hold K=2,3
    if (lane < 16) { b.x = -2.0f * qx; b.y = -2.0f * qy;        }
    else           { b.x = 1.0f;       b.y = qx * qx + qy * qy; }

    unsigned s0 = 0xFFFFFFFFu, s1 = 0xFFFFFFFFu, s2 = 0xFFFFFFFFu, s3 = 0xFFFFFFFFu;
    const unsigned hiOff = (lane >= 16) ? 8u : 0u;     // D rows M: lo lanes 0-7, hi lanes 8-15
    const v2f* aPtr = aStage + lane;

    v2f aCur = aPtr[0];                                // prologue fetch (tile 0)
    for (int t = 0; t < NTILES; ++t) {
        // software pipeline: issue next tile's LDS load before the VALU work
        const int tn = (t + 1 < NTILES) ? (t + 1) : t;
        const v2f aNext = aPtr[tn * 32];               // ds_load_b64, waits overlap top-4 net

        v8f d = {};
        d = __builtin_amdgcn_wmma_f32_16x16x4_f32(
                /*neg_a=*/false, aCur, /*neg_b=*/false, b,
                /*c_mod=*/(short)0, d, /*reuse_a=*/false, /*reuse_b=*/false);

        const unsigned base = (unsigned)(t * 16) + hiOff;
        #pragma unroll
        for (int r = 0; r < 8; ++r) {
            // clamp tiny cancellation negatives with a single v_med3_f32
            const float dr = __builtin_amdgcn_fmed3f(d[r], 0.0f, FLT_BIG);
            const unsigned key = (__float_as_uint(dr) & 0xFFFFE000u) | (base + (unsigned)r);
            TOP4_INSERT(key);
        }
        aCur = aNext;
    }

    // merge the two half-lists for each query (lane L <-> lane L+16)
    {
        const unsigned p0 = __shfl_xor(s0, 16, 32);
        const unsigned p1 = __shfl_xor(s1, 16, 32);
        const unsigned p2 = __shfl_xor(s2, 16, 32);
        const unsigned p3 = __shfl_xor(s3, 16, 32);
        TOP4_INSERT(p0);
        TOP4_INSERT(p1);
        TOP4_INSERT(p2);
        TOP4_INSERT(p3);
    }

    if (lane < 16) {
        const int qi = wave * 16 + lane;               // block-local query 0..127
        idxBuf[qi * 4 + 0] = (int)(s0 & 0x1FFFu);
        idxBuf[qi * 4 + 1] = (int)(s1 & 0x1FFFu);
        idxBuf[qi * 4 + 2] = (int)(s2 & 0x1FFFu);
        idxBuf[qi * 4 + 3] = (int)(s3 & 0x1FFFu);
    }
    __syncthreads();

    // ---------------- Phase 3: gather 4 neighbors, mean over k --------------
    {
        const int qi = tid & (QPB - 1);
        const int q  = qbase + qi;
        const int i0 = idxBuf[qi * 4 + 0];
        const int i1 = idxBuf[qi * 4 + 1];
        const int i2 = idxBuf[qi * 4 + 2];
        const int i3 = idxBuf[qi * 4 + 3];
        for (int c = (tid >> 7); c < C_CH; c += 2) {
            const float* vr = values + (size_t)c * N_IN;
            const float s = vr[i0] + vr[i1] + vr[i2] + vr[i3];
            out[(size_t)c * N_TOT + q] = 0.25f * s;   // coalesced over 128 queries
        }
    }
}

extern "C" void kernel_launch(void* const* d_in, const int* in_sizes, int n_in,
                              void* d_out, int out_size, void* d_ws, size_t ws_size,
                              hipStream_t stream) {
    (void)in_sizes; (void)n_in; (void)out_size; (void)d_ws; (void)ws_size;
    const float* values     = (const float*)d_in[0];   // [64, 8192]
    const float* coords     = (const float*)d_in[1];   // [8192, 2]
    const float* new_coords = (const float*)d_in[2];   // [24576, 2]
    const float* shift      = (const float*)d_in[3];   // [2]
    float* out = (float*)d_out;                        // [64, 32768]

    hipLaunchKernelGGL(knn_upsample_wmma,
                       dim3(N_TOT / QPB), dim3(THREADS), LDS_BYTES, stream,
                       values, coords, new_coords, shift, out);
}